// mimoaf_23106924052731
// MI455X (gfx1250) — compile-verified
//
#include <hip/hip_runtime.h>
#include <math.h>

// ---------------------------------------------------------------------------
// DD-LMS MIMO equalizer scan for gfx1250 (MI455X).
// Serial scan (65537 dependent steps) -> latency-bound. One wave32, lane=tap.
// Cross-lane reduction: two chained V_WMMA_F32_16X16X4_F32 ops with disjoint
// column-selection B matrices route the 4 reduction channels (re/im x 2 pol)
// into columns 0..3 of a single accumulator; finished with one 8-VGPR sum and
// v_readlane broadcasts (no LDS, no DS waits in the serial loop).
// ---------------------------------------------------------------------------

typedef __attribute__((ext_vector_type(2))) float v2f;
typedef __attribute__((ext_vector_type(8))) float v8f;

#define TAPS    32
#define NTOT    131104
#define NSTEPS  65537          // (131104 - 32)/2 + 1
#define GMAX    30.0f
#define SQRT10  3.1622776601683795f
#define ISQRT10 0.31622776601683794f

struct cx { float x, y; };

__device__ __forceinline__ cx cmul (cx a, cx b){ return { a.x*b.x - a.y*b.y, a.x*b.y + a.y*b.x }; }
__device__ __forceinline__ cx cmulc(cx a, cx b){ return { a.x*b.x + a.y*b.y, a.y*b.x - a.x*b.y }; } // a*conj(b)
__device__ __forceinline__ cx cadd (cx a, cx b){ return { a.x + b.x, a.y + b.y }; }
__device__ __forceinline__ cx csub (cx a, cx b){ return { a.x - b.x, a.y - b.y }; }
__device__ __forceinline__ cx cscale(cx a, float s){ return { a.x*s, a.y*s }; }

__device__ __forceinline__ cx clip30(cx g){
    float m  = sqrtf(g.x*g.x + g.y*g.y);
    float sc = GMAX / fmaxf(m, GMAX);
    return { g.x*sc, g.y*sc };
}

// nearest of {-3,-1,1,3} (16-QAM is separable per axis; matches argmin up to
// measure-zero ties)
__device__ __forceinline__ float slice1(float t){
    float p = 2.0f*rintf(0.5f*(t - 1.0f)) + 1.0f;
    return fminf(3.0f, fmaxf(-3.0f, p));
}
__device__ __forceinline__ cx qamslice(cx q){
    return { slice1(q.x*SQRT10)*ISQRT10, slice1(q.y*SQRT10)*ISQRT10 };
}

// uniform broadcast read of one lane (v_readlane_b32 -> SGPR, no LDS)
__device__ __forceinline__ float rlane(float v, int l){
    return __int_as_float(__builtin_amdgcn_readlane(__float_as_int(v), l));
}

__global__ __launch_bounds__(32) void ddlms_scan_kernel(
    const float* __restrict__ xre, const float* __restrict__ xim,
    const float* __restrict__ p_ew, const float* __restrict__ p_ef,
    const float* __restrict__ p_es, const float* __restrict__ p_eb,
    const float* __restrict__ p_bp, float* __restrict__ y)
{
    const int lane = threadIdx.x;          // tap index 0..31

    const float lr_w = tanhf(p_ew[0]);
    const float lr_f = expf (p_ef[0]);
    const float lr_s = expf (p_es[0]);
    const float lr_b = expf (p_eb[0]);
    const float beta = expf (p_bp[0]);
    const float omb  = 1.0f - beta;

    // Disjoint column-selection B matrices (constant over the loop).
    // B layout (4x16 f32): VGPR0 = rows k=0 (lanes 0-15) / k=2 (lanes 16-31),
    //                      VGPR1 = rows k=1 / k=3.
    // bselA: col 0 <- k{0,2} (pol0.re pairsums), col 1 <- k{1,3} (pol0.im)
    // bselB: col 2 <- k{0,2} (pol1.re),          col 3 <- k{1,3} (pol1.im)
    const int col = lane & 15;
    v2f bselA, bselB;
    bselA.x = (col == 0) ? 1.0f : 0.0f;
    bselA.y = (col == 1) ? 1.0f : 0.0f;
    bselB.x = (col == 2) ? 1.0f : 0.0f;
    bselB.y = (col == 3) ? 1.0f : 0.0f;

    // carry: w[i][j][lane] per lane, scalars replicated on all lanes
    cx w00 = {0,0}, w01 = {0,0}, w10 = {0,0}, w11 = {0,0};
    if (lane == TAPS/2) { w00.x = 1.0f; w11.x = 1.0f; }
    cx f0={1,0}, f1={1,0}, s0={1,0}, s1={1,0};
    cx b0={0,0}, b1={0,0}, fh0={1,0}, fh1={1,0};

    // software-pipelined sliding-window loads: u[t][j] = x[2n+t][j]
    int row = lane;
    float2 ur = *(const float2*)(xre + 2*row);
    float2 ui = *(const float2*)(xim + 2*row);

    for (int n = 0; n < NSTEPS; ++n) {
        // prefetch step n+1 (clamped; overlaps the compute below)
        int nrow = 2*(n+1) + lane; if (nrow > NTOT-1) nrow = NTOT-1;
        float2 nur = *(const float2*)(xre + 2*nrow);
        float2 nui = *(const float2*)(xim + 2*nrow);

        cx u0 = { ur.x, ui.x };
        cx u1 = { ur.y, ui.y };

        // per-lane tap partials: v_i = sum_t sum_j w[i][j][t]*u[t][j]
        cx pa = cadd(cmul(w00, u0), cmul(w01, u1));   // output 0
        cx pb = cadd(cmul(w10, u0), cmul(w11, u1));   // output 1

        // chained WMMA reduction: D[m][c] = pairsum(channel c)[m], c=0..3
        v2f aA; aA.x = pa.x; aA.y = pa.y;
        v2f aB; aB.x = pb.x; aB.y = pb.y;
        v8f acc = {};
        acc = __builtin_amdgcn_wmma_f32_16x16x4_f32(
            false, aA, false, bselA, (short)0, acc, false, false);
        acc = __builtin_amdgcn_wmma_f32_16x16x4_f32(
            false, aB, false, bselB, (short)0, acc, false, false);

        // lane L holds sum over its half of D[m][L%16]; combine lanes c, c+16
        float s8 = ((acc[0]+acc[1])+(acc[2]+acc[3]))
                 + ((acc[4]+acc[5])+(acc[6]+acc[7]));
        cx v0 = { rlane(s8, 0) + rlane(s8, 16), rlane(s8, 1) + rlane(s8, 17) };
        cx v1 = { rlane(s8, 2) + rlane(s8, 18), rlane(s8, 3) + rlane(s8, 19) };

        // scalar tail (uniform across lanes)
        cx k0 = cmul(v0, f0),  k1 = cmul(v1, f1);
        cx c0 = cmul(k0, s0),  c1 = cmul(k1, s1);
        cx z0 = cadd(c0, b0),  z1 = cadd(c1, b1);
        cx q0 = cadd(cmul(v0, fh0), b0);
        cx q1 = cadd(cmul(v1, fh1), b1);
        cx d0 = qamslice(q0),  d1 = qamslice(q1);

        cx fs0 = cmul(f0, s0), fs1 = cmul(f1, s1);
        float i0 = 1.0f / sqrtf(fs0.x*fs0.x + fs0.y*fs0.y);
        float i1 = 1.0f / sqrtf(fs1.x*fs1.x + fs1.y*fs1.y);
        cx psi0 = { fs0.x*i0, -fs0.y*i0 };            // conj(f*s)/|f*s|
        cx psi1 = { fs1.x*i1, -fs1.y*i1 };

        cx db0 = csub(d0, b0), db1 = csub(d1, b1);
        cx ew0 = csub(cmul(db0, psi0), v0);
        cx ew1 = csub(cmul(db1, psi1), v1);
        cx ef0 = csub(db0, k0), ef1 = csub(db1, k1);
        cx es0 = csub(db0, c0), es1 = csub(db1, c1);
        cx eb0 = csub(d0, z0),  eb1 = csub(d1, z1);

        // y[n] = z (pre-update state) : complex64 interleaved (n, 2)
        if (lane == 0) {
            float4 o; o.x = z0.x; o.y = z0.y; o.z = z1.x; o.w = z1.y;
            *(float4*)(y + 4*n) = o;
        }

        // updates (w -= lr*clip(-e*conj(.)) == w += lr*clip(e*conj(.)))
        f0 = cadd(f0, cscale(clip30(cmulc(ef0, k0)), lr_f));
        f1 = cadd(f1, cscale(clip30(cmulc(ef1, k1)), lr_f));
        s0 = cadd(s0, cscale(cmulc(es0, c0), lr_s));
        s1 = cadd(s1, cscale(cmulc(es1, c1), lr_s));
        b0 = cadd(b0, cscale(eb0, lr_b));
        b1 = cadd(b1, cscale(eb1, lr_b));

        w00 = cadd(w00, cscale(clip30(cmulc(ew0, u0)), lr_w));
        w01 = cadd(w01, cscale(clip30(cmulc(ew0, u1)), lr_w));
        w10 = cadd(w10, cscale(clip30(cmulc(ew1, u0)), lr_w));
        w11 = cadd(w11, cscale(clip30(cmulc(ew1, u1)), lr_w));

        // fshat uses post-update f, s
        fh0 = cadd(cscale(fh0, beta), cscale(cmul(f0, s0), omb));
        fh1 = cadd(cscale(fh1, beta), cscale(cmul(f1, s1), omb));

        ur = nur; ui = nui;
    }
}

extern "C" void kernel_launch(void* const* d_in, const int* in_sizes, int n_in,
                              void* d_out, int out_size, void* d_ws, size_t ws_size,
                              hipStream_t stream) {
    (void)in_sizes; (void)n_in; (void)out_size; (void)d_ws; (void)ws_size;
    const float* xre = (const float*)d_in[0];
    const float* xim = (const float*)d_in[1];
    const float* ew  = (const float*)d_in[2];
    const float* ef  = (const float*)d_in[3];
    const float* es  = (const float*)d_in[4];
    const float* eb  = (const float*)d_in[5];
    const float* bp  = (const float*)d_in[6];

    // inherently serial scan: one wave32 owns the whole carry in registers
    ddlms_scan_kernel<<<dim3(1), dim3(32), 0, stream>>>(
        xre, xim, ew, ef, es, eb, bp, (float*)d_out);
}